// Model_16003048145508
// MI455X (gfx1250) — compile-verified
//
#include <hip/hip_runtime.h>

typedef __attribute__((ext_vector_type(2))) float v2f;
typedef __attribute__((ext_vector_type(4))) float v4f;
typedef __attribute__((ext_vector_type(8))) float v8f;

#define K_FEAT 41024
#define KCHUNK 512
#define NK ((K_FEAT + KCHUNK - 1) / KCHUNK)   // 81 (80 full chunks + one of 64)

// GEMV via V_WMMA_F32_16X16X4_F32, split-K.
// grid.x = NK*4 blocks, 256 threads (8 waves). Each wave = one (row-tile, k-chunk):
//   tile 0..15  -> rows of W_my  (outputs 0..255)
//   tile 16..31 -> rows of W_opp (outputs 256..511)
// A = W tile (16 rows x 4 K), documented f32 A layout:
//   lane L: M = L%16, holds K = 2*(L/16) + {0,1}  -> one b64 load per lane.
// B = x broadcast over all 16 N columns (B[k][n] = x[k]) -> every D column
//   equals the GEMV partial; we read out column N=0 (lane 0: M=0..7, lane 16: M=8..15).
__global__ __launch_bounds__(256) void nnue_gemv_wmma(
    const float* __restrict__ x,
    const float* __restrict__ Wmy,
    const float* __restrict__ Wopp,
    float* __restrict__ partial) {
  const int blk   = blockIdx.x;
  const int chunk = blk >> 2;                 // 0..NK-1
  const int tgrp  = blk & 3;                  // 0..3
  const int wave  = threadIdx.x >> 5;         // 0..7
  const int lane  = threadIdx.x & 31;
  const int tile  = tgrp * 8 + wave;          // 0..31
  const int side  = tile >> 4;                // 0 = my, 1 = opp
  const int mrow  = (tile & 15) * 16 + (lane & 15);
  const int koff  = (lane >> 4) * 2;          // 0 or 2

  const float* W  = side ? Wopp : Wmy;
  const float* wp = W + (size_t)mrow * K_FEAT + koff;
  const float* xp = x + side * K_FEAT + koff;

  const int k0   = chunk * KCHUNK;
  const int kend = (k0 + KCHUNK < K_FEAT) ? (k0 + KCHUNK) : K_FEAT;
  // every chunk length is a multiple of 32 (K_FEAT % 32 == 0, KCHUNK % 32 == 0)

  v8f c = {};
  for (int k = k0; k < kend; k += 32) {
    __builtin_prefetch(wp + k + 256, 0, 0);   // global_prefetch_b8, speculative
#pragma unroll
    for (int kk = 0; kk < 32; kk += 4) {
      v2f a = *(const v2f*)(wp + k + kk);     // 8B-aligned b64 load of W
      v2f b = *(const v2f*)(xp + k + kk);     // x chunk, L2-resident
      c = __builtin_amdgcn_wmma_f32_16x16x4_f32(
          /*neg_a=*/false, a, /*neg_b=*/false, b,
          /*c_mod=*/(short)0, c, /*reuse_a=*/false, /*reuse_b=*/false);
    }
  }

  // Column N=0 of D lives in lane 0 (M=0..7 in c[0..7]) and lane 16 (M=8..15).
  if ((lane & 15) == 0) {
    float* o = partial + (size_t)chunk * 512 + tile * 16 + (lane >> 4) * 8;
    v4f lo = {c[0], c[1], c[2], c[3]};
    v4f hi = {c[4], c[5], c[6], c[7]};
    *(v4f*)(o)     = lo;
    *(v4f*)(o + 4) = hi;
  }
}

// Reduce split-K partials + bias + ReLU, then the tiny 512->32->32->1 MLP.
__global__ __launch_bounds__(512) void nnue_tail(
    const float* __restrict__ partial,
    const float* __restrict__ bmy, const float* __restrict__ bopp,
    const float* __restrict__ W1, const float* __restrict__ b1,
    const float* __restrict__ W2, const float* __restrict__ b2,
    const float* __restrict__ W3, const float* __restrict__ b3,
    float* __restrict__ out) {
  __shared__ float h[512];
  __shared__ float h1[32];
  __shared__ float h2[32];
  const int t = threadIdx.x;

  float s = (t < 256) ? bmy[t] : bopp[t - 256];
  for (int c = 0; c < NK; ++c) s += partial[(size_t)c * 512 + t];
  h[t] = s > 0.f ? s : 0.f;
  __syncthreads();

  if (t < 32) {
    float a = b1[t];
    const float* w = W1 + t * 512;
    for (int i = 0; i < 512; ++i) a += w[i] * h[i];
    h1[t] = a > 0.f ? a : 0.f;
  }
  __syncthreads();

  if (t < 32) {
    float a = b2[t];
    const float* w = W2 + t * 32;
    for (int i = 0; i < 32; ++i) a += w[i] * h1[i];
    h2[t] = a > 0.f ? a : 0.f;
  }
  __syncthreads();

  if (t == 0) {
    float a = b3[0];
    for (int i = 0; i < 32; ++i) a += W3[i] * h2[i];
    out[0] = a;
  }
}

extern "C" void kernel_launch(void* const* d_in, const int* in_sizes, int n_in,
                              void* d_out, int out_size, void* d_ws, size_t ws_size,
                              hipStream_t stream) {
  (void)in_sizes; (void)n_in; (void)out_size; (void)ws_size;
  const float* x    = (const float*)d_in[0];
  const float* Wmy  = (const float*)d_in[1];
  const float* bmy  = (const float*)d_in[2];
  const float* Wopp = (const float*)d_in[3];
  const float* bopp = (const float*)d_in[4];
  const float* W1   = (const float*)d_in[5];
  const float* b1   = (const float*)d_in[6];
  const float* W2   = (const float*)d_in[7];
  const float* b2   = (const float*)d_in[8];
  const float* W3   = (const float*)d_in[9];
  const float* b3   = (const float*)d_in[10];
  float* out     = (float*)d_out;
  float* partial = (float*)d_ws;      // NK * 512 floats = 166 KB

  nnue_gemv_wmma<<<dim3(NK * 4), dim3(256), 0, stream>>>(x, Wmy, Wopp, partial);
  nnue_tail<<<dim3(1), dim3(512), 0, stream>>>(partial, bmy, bopp,
                                               W1, b1, W2, b2, W3, b3, out);
}